// SelfAttention_40699110097492
// MI455X (gfx1250) — compile-verified
//
#include <hip/hip_runtime.h>
#include <hip/hip_bf16.h>

// Self-attention (bs=2, seq=2048, d=1024, h=16, hd=64) for MI455X (gfx1250).
// Flash-attention structure: scores never touch HBM. All matmuls use the
// native fp32 WMMA V_WMMA_F32_16X16X4_F32 (exact vs fp32 reference).
// Workspace: Q,K,V in [b,h,s,64] layout = 3 * 16 MB = 48 MB in d_ws.

typedef __attribute__((ext_vector_type(2))) float v2f;
typedef __attribute__((ext_vector_type(8))) float v8f;

#define BS   2
#define SEQ  2048
#define DMODEL 1024
#define HEADS 16
#define HD   64
#define N3   3072
#define MTOT (BS * SEQ)            // 4096

__device__ __forceinline__ v8f wmma_f32_16x16x4(v2f a, v2f b, v8f c) {
  // 8 args: (neg_a, A, neg_b, B, c_mod, C, reuse_a, reuse_b)
  return __builtin_amdgcn_wmma_f32_16x16x4_f32(false, a, false, b, (short)0, c,
                                               false, false);
}

// ---------------------------------------------------------------------------
// Kernel 1: QKV projection.  One wave computes a 64(M) x 64(N) tile of
// E[4096,1024] @ W[1024,3072] + bias, then scatters into head-split layout.
// Reference head split: q[b,s,:] viewed as (hd, h) -> head = col%16, dim=col/16.
// Q is pre-scaled by 1/sqrt(hd) = 0.125.
// ---------------------------------------------------------------------------
__global__ void __launch_bounds__(32)
qkv_proj_kernel(const float* __restrict__ E, const float* __restrict__ W,
                const float* __restrict__ bias, float* __restrict__ Qw,
                float* __restrict__ Kw, float* __restrict__ Vw) {
  const int lane  = threadIdx.x & 31;
  const int n15   = lane & 15;          // N (or M-row for A loads) within tile
  const int khalf = (lane >> 4) << 1;   // 0 or 2 : K offset for this lane half
  const int NT    = N3 / 64;            // 48 tiles along N
  const int mt    = blockIdx.x / NT;    // 0..63
  const int nt    = blockIdx.x % NT;    // 0..47
  const int nbase = nt * 64;

  v8f acc[4][4];
#pragma unroll
  for (int mm = 0; mm < 4; ++mm)
#pragma unroll
    for (int t = 0; t < 4; ++t) acc[mm][t] = v8f{};

  for (int k4 = 0; k4 < DMODEL / 4; ++k4) {   // 256 K-steps of 4
    const int kb = k4 * 4 + khalf;
    v2f a[4];
#pragma unroll
    for (int mm = 0; mm < 4; ++mm) {
      const float* ap =
          E + (size_t)(mt * 64 + mm * 16 + n15) * DMODEL + kb;
      a[mm] = *(const v2f*)ap;                 // global_load_b64
    }
#pragma unroll
    for (int t = 0; t < 4; ++t) {
      const int ncol = nbase + t * 16 + n15;
      v2f bf;
      bf.x = W[(size_t)kb * N3 + ncol];
      bf.y = W[(size_t)(kb + 1) * N3 + ncol];
#pragma unroll
      for (int mm = 0; mm < 4; ++mm)
        acc[mm][t] = wmma_f32_16x16x4(a[mm], bf, acc[mm][t]);
    }
  }

  // Epilogue: bias add + scatter to [b, head, s, dim] (Q scaled by 0.125).
  const int mbase = (lane >> 4) * 8;
#pragma unroll
  for (int t = 0; t < 4; ++t) {
    const int ncol  = nbase + t * 16 + n15;
    const float bv  = bias[ncol];
    const int which = ncol >> 10;        // 0=q 1=k 2=v
    const int c     = ncol & 1023;
    const int head  = c & 15;
    const int dim   = c >> 4;
    float* dst = (which == 0) ? Qw : (which == 1) ? Kw : Vw;
    const float scale = (which == 0) ? 0.125f : 1.0f;
#pragma unroll
    for (int mm = 0; mm < 4; ++mm) {
#pragma unroll
      for (int r = 0; r < 8; ++r) {
        const int mrow = mt * 64 + mm * 16 + mbase + r;
        const int b = mrow >> 11;        // /2048
        const int s = mrow & 2047;
        dst[(((size_t)(b * HEADS + head)) * SEQ + s) * HD + dim] =
            (acc[mm][t][r] + bv) * scale;
      }
    }
  }
}

// ---------------------------------------------------------------------------
// Kernel 2: flash attention. One wave per (b, h, 32-query tile): two 16-row
// M-subtiles share every K and V fragment (2 WMMAs per K-load, 2 per V-pair).
// Per 16-key chunk:
//   S = Q @ K^T  (2x16 WMMAs) -> online softmax (shfl_xor row reductions)
//   P bounced through LDS (D-layout -> A-layout), O += P @ V (2x16 WMMAs).
// Next chunk's K/V prefetched via global_prefetch.
// ---------------------------------------------------------------------------
__global__ void __launch_bounds__(32)
flash_attn_kernel(const float* __restrict__ Qw, const float* __restrict__ Kw,
                  const float* __restrict__ Vw, float* __restrict__ out) {
  __shared__ float pbuf[2][16 * 17];    // padded 16x16 P tiles (single wave)

  const int lane  = threadIdx.x & 31;
  const int n15   = lane & 15;
  const int khalf = (lane >> 4) << 1;
  const int mbase = (lane >> 4) * 8;

  const int tile = blockIdx.x;          // 0..2047
  const int qt = tile & 63;             // 32-row query tile in seq
  const int bh = tile >> 6;             // b*16 + h
  const int h  = bh & 15;
  const int b  = bh >> 4;

  const float* Qh = Qw + (((size_t)bh) * SEQ + qt * 32) * HD;
  const float* Kh = Kw + ((size_t)bh) * SEQ * HD;
  const float* Vh = Vw + ((size_t)bh) * SEQ * HD;

  // Q fragments: A-layout, 16 K-steps covering hd=64, for 2 M-subtiles.
  v2f qa[2][16];
#pragma unroll
  for (int u = 0; u < 2; ++u)
#pragma unroll
    for (int kk = 0; kk < 16; ++kk)
      qa[u][kk] = *(const v2f*)(Qh + (u * 16 + n15) * HD + kk * 4 + khalf);

  v8f acc[2][4];
#pragma unroll
  for (int u = 0; u < 2; ++u)
#pragma unroll
    for (int t = 0; t < 4; ++t) acc[u][t] = v8f{};
  float mi[2][8], li[2][8];
#pragma unroll
  for (int u = 0; u < 2; ++u)
#pragma unroll
    for (int r = 0; r < 8; ++r) { mi[u][r] = -3.0e38f; li[u][r] = 0.0f; }

  for (int kc = 0; kc < SEQ / 16; ++kc) {        // 128 key chunks
    // ---- prefetch next chunk's K and V tiles (4 KB each) -------------
    if (kc + 1 < SEQ / 16) {
      const float* Kn = Kh + (size_t)(kc + 1) * 16 * HD;
      const float* Vn = Vh + (size_t)(kc + 1) * 16 * HD;
      __builtin_prefetch(Kn + lane * 32, 0, 1);  // 32 lanes x 128 B = 4 KB
      __builtin_prefetch(Vn + lane * 32, 0, 1);
    }

    // ---- S = Q @ K^T for 16 keys, both M-subtiles --------------------
    const float* Kc = Kh + (size_t)(kc * 16 + n15) * HD;
    v8f s0 = v8f{}, s1 = v8f{};
#pragma unroll
    for (int kk = 0; kk < 16; ++kk) {
      v2f bf = *(const v2f*)(Kc + kk * 4 + khalf); // B[k][n] = K[n][k]
      s0 = wmma_f32_16x16x4(qa[0][kk], bf, s0);
      s1 = wmma_f32_16x16x4(qa[1][kk], bf, s1);
    }

    // ---- online softmax (both subtiles) ------------------------------
    float alpha[2][8];
    v8f p[2];
#pragma unroll
    for (int u = 0; u < 2; ++u) {
      v8f s = (u == 0) ? s0 : s1;
#pragma unroll
      for (int r = 0; r < 8; ++r) {
        float rowmax = s[r];                     // reduce over N (16 lanes)
        rowmax = fmaxf(rowmax, __shfl_xor(rowmax, 1, 32));
        rowmax = fmaxf(rowmax, __shfl_xor(rowmax, 2, 32));
        rowmax = fmaxf(rowmax, __shfl_xor(rowmax, 4, 32));
        rowmax = fmaxf(rowmax, __shfl_xor(rowmax, 8, 32));
        const float newm = fmaxf(mi[u][r], rowmax);
        alpha[u][r] = __expf(mi[u][r] - newm);
        mi[u][r] = newm;
        const float pv = __expf(s[r] - newm);
        p[u][r] = pv;
        float rs = pv;
        rs += __shfl_xor(rs, 1, 32);
        rs += __shfl_xor(rs, 2, 32);
        rs += __shfl_xor(rs, 4, 32);
        rs += __shfl_xor(rs, 8, 32);
        li[u][r] = li[u][r] * alpha[u][r] + rs;
      }
#pragma unroll
      for (int t = 0; t < 4; ++t)
#pragma unroll
        for (int r = 0; r < 8; ++r) acc[u][t][r] *= alpha[u][r];
    }

    // ---- P: D-layout -> LDS -> A-layout ------------------------------
#pragma unroll
    for (int u = 0; u < 2; ++u)
#pragma unroll
      for (int r = 0; r < 8; ++r)
        pbuf[u][(mbase + r) * 17 + n15] = p[u][r];
    __syncthreads();                   // single-wave WG: NOP + compiler fence

    // ---- O += P @ V (V fragments shared by both subtiles) ------------
    const float* Vc = Vh + (size_t)(kc * 16) * HD;
#pragma unroll
    for (int kk2 = 0; kk2 < 4; ++kk2) {
      const int k0 = kk2 * 4 + khalf;
      v2f af[2];
#pragma unroll
      for (int u = 0; u < 2; ++u) {
        af[u].x = pbuf[u][n15 * 17 + k0];
        af[u].y = pbuf[u][n15 * 17 + k0 + 1];
      }
#pragma unroll
      for (int t = 0; t < 4; ++t) {
        v2f bf;
        bf.x = Vc[(size_t)k0 * HD + t * 16 + n15];
        bf.y = Vc[(size_t)(k0 + 1) * HD + t * 16 + n15];
        acc[0][t] = wmma_f32_16x16x4(af[0], bf, acc[0][t]);
        acc[1][t] = wmma_f32_16x16x4(af[1], bf, acc[1][t]);
      }
    }
    __syncthreads();
  }

  // ---- epilogue: normalize and store out[b, s, h*64 + col] -------------
#pragma unroll
  for (int u = 0; u < 2; ++u) {
#pragma unroll
    for (int t = 0; t < 4; ++t) {
#pragma unroll
      for (int r = 0; r < 8; ++r) {
        const int srow = qt * 32 + u * 16 + mbase + r;
        out[(((size_t)b * SEQ) + srow) * DMODEL + h * HD + t * 16 + n15] =
            acc[u][t][r] / li[u][r];
      }
    }
  }
}

extern "C" void kernel_launch(void* const* d_in, const int* in_sizes, int n_in,
                              void* d_out, int out_size, void* d_ws,
                              size_t ws_size, hipStream_t stream) {
  const float* E    = (const float*)d_in[0];   // [2,2048,1024]
  const float* W    = (const float*)d_in[1];   // [1024,3072]
  const float* bias = (const float*)d_in[2];   // [3072]
  float* out = (float*)d_out;                  // [2,2048,1024]

  const size_t per = (size_t)BS * HEADS * SEQ * HD; // 4 Mi floats = 16 MB
  float* Qw = (float*)d_ws;
  float* Kw = Qw + per;
  float* Vw = Kw + per;

  // 64x64 tiles over (M=4096, N=3072): 64*48 waves, one wave per block.
  qkv_proj_kernel<<<(MTOT / 64) * (N3 / 64), 32, 0, stream>>>(E, W, bias, Qw,
                                                              Kw, Vw);
  // one wave per (b, h, 32-query tile): 2*16*64 = 2048 blocks.
  flash_attn_kernel<<<BS * HEADS * (SEQ / 32), 32, 0, stream>>>(Qw, Kw, Vw,
                                                                out);
}